// Net_59416577572910
// MI455X (gfx1250) — compile-verified
//
#include <hip/hip_runtime.h>
#include <hip/hip_bf16.h>

#define K_TAGS   16
#define T_LEN    512
#define B_BATCH  1024
#define PAD_ID   0
#define START_ID 14
#define STOP_ID  15

typedef __attribute__((ext_vector_type(2))) float v2f;
typedef __attribute__((ext_vector_type(8))) float v8f;

// Swap lane halves (lane ^ 16) via ds_swizzle SWAPX16: xor=0x10, and=0x1f -> imm 0x401F
__device__ __forceinline__ float swap16(float x) {
  return __int_as_float(__builtin_amdgcn_ds_swizzle(__float_as_int(x), 0x401F));
}

// ---------------------------------------------------------------------------
// Forward score. One wave32 per tile of 16 batch rows.
// State P: scaled probabilities exp(alpha)*2^-etot in WMMA C/D layout:
//   VGPR v, lane l -> tag j = v + 8*(l/16), batch b = l%16
// Step: Q = E x P  (E = exp(trans), A operand, loop invariant),
//       P' = exp(emit) .* Q, then exact power-of-two renormalization with the
//       scale tracked as an integer exponent (no divide / log on the chain).
// ---------------------------------------------------------------------------
__global__ __launch_bounds__(32) void crf_forward_kernel(
    const float* __restrict__ h, const float* __restrict__ mask,
    const float* __restrict__ trans, float* __restrict__ fwd_out)
{
  const int lane = threadIdx.x;       // 0..31
  const int half = lane >> 4;         // lane half
  const int bl   = lane & 15;
  const int brow = blockIdx.x * K_TAGS + bl;   // global batch row (N index)

  // A = exp(trans) in A layout: chunk c, component v: lane holds
  //   M = j = lane%16,  K = i = 4c + v + 2*(lane/16)
  v2f A[4];
#pragma unroll
  for (int c = 0; c < 4; ++c) {
    A[c].x = __expf(trans[bl * K_TAGS + (4 * c + 0 + 2 * half)]);
    A[c].y = __expf(trans[bl * K_TAGS + (4 * c + 1 + 2 * half)]);
  }
  // Stop weights exp(trans[STOP, j]) in C/D layout
  float estop[8];
#pragma unroll
  for (int v = 0; v < 8; ++v)
    estop[v] = __expf(trans[STOP_ID * K_TAGS + (v + 8 * half)]);

  // alpha0 = NEG except START=0  ->  p = one-hot(START), scale exponent 0
  float P[8];
#pragma unroll
  for (int v = 0; v < 8; ++v) P[v] = ((v + 8 * half) == START_ID) ? 1.0f : 0.0f;
  int etot = 0;                        // exact accumulated log2 scale

  const float* hp = h + (size_t)brow * T_LEN * K_TAGS + 8 * half; // lane's emit stream
  const float* mp = mask + (size_t)brow * T_LEN;

  for (int t = 0; t < T_LEN; ++t) {
    if (t + 8 < T_LEN)
      __builtin_prefetch(hp + (size_t)(t + 8) * K_TAGS, 0, 3); // global_prefetch_b8

    const float4 e0 = *(const float4*)(hp + (size_t)t * K_TAGS);
    const float4 e1 = *(const float4*)(hp + (size_t)t * K_TAGS + 4);
    const float  mt = mp[t];

    // Rebuild B operand (K=i in VGPR dim, b in lanes) from P (j in VGPR dim):
    // half-wave swaps + per-half selects, no LDS memory traffic.
    const float s0 = swap16(P[0]), s1 = swap16(P[1]), s2 = swap16(P[2]), s3 = swap16(P[3]);
    const float s4 = swap16(P[4]), s5 = swap16(P[5]), s6 = swap16(P[6]), s7 = swap16(P[7]);
    v2f Bm[4];
    Bm[0].x = half ? s2   : P[0];  Bm[0].y = half ? s3   : P[1];  // i = 0,1 | 2,3
    Bm[1].x = half ? s6   : P[4];  Bm[1].y = half ? s7   : P[5];  // i = 4,5 | 6,7
    Bm[2].x = half ? P[2] : s0;    Bm[2].y = half ? P[3] : s1;    // i = 8,9 |10,11
    Bm[3].x = half ? P[6] : s4;    Bm[3].y = half ? P[7] : s5;    // i =12,13|14,15

    // Q = E x P : four chained f32 WMMAs, K=4 each, accumulate through C.
    v8f acc = {};
#pragma unroll
    for (int c = 0; c < 4; ++c)
      acc = __builtin_amdgcn_wmma_f32_16x16x4_f32(
          false, A[c], false, Bm[c], (short)0, acc, false, false);

    // p' = exp(emit) .* Q ; masked rows keep old state (emit exp is off the chain)
    const bool on = mt > 0.5f;
    float pn[8];
    pn[0] = on ? acc[0] * __expf(e0.x) : P[0];
    pn[1] = on ? acc[1] * __expf(e0.y) : P[1];
    pn[2] = on ? acc[2] * __expf(e0.z) : P[2];
    pn[3] = on ? acc[3] * __expf(e0.w) : P[3];
    pn[4] = on ? acc[4] * __expf(e1.x) : P[4];
    pn[5] = on ? acc[5] * __expf(e1.y) : P[5];
    pn[6] = on ? acc[6] * __expf(e1.z) : P[6];
    pn[7] = on ? acc[7] * __expf(e1.w) : P[7];

    // Exact power-of-two renormalization: m = f * 2^e, f in [0.5,1).
    // Scale by 2^-e (exact multiply), accumulate e as an integer (exact).
    // Masked rows: max(P) in [0.5,1) -> e = 0 -> exact no-op.
    float m = pn[0];
#pragma unroll
    for (int v = 1; v < 8; ++v) m = fmaxf(m, pn[v]);
    m = fmaxf(m, swap16(m));          // combine j<8 and j>=8 halves
    const int   e = (__float_as_int(m) >> 23) - 126;     // m always positive normal
    const float r = __int_as_float((127 - e) << 23);     // 2^-e, exact
#pragma unroll
    for (int v = 0; v < 8; ++v) P[v] = pn[v] * r;
    etot += e;
  }

  // fwd = etot*ln2 + log( sum_j P[j] * exp(trans[STOP, j]) )
  float s = 0.0f;
#pragma unroll
  for (int v = 0; v < 8; ++v) s += P[v] * estop[v];
  s += swap16(s);
  const float fwd =
      (float)((double)etot * 0.6931471805599453 + (double)__logf(s));
  if (lane < 16) fwd_out[brow] = fwd;
}

// ---------------------------------------------------------------------------
// Gold score: one wave per sequence, lanes stride over T.
// ---------------------------------------------------------------------------
__global__ __launch_bounds__(256) void crf_gold_kernel(
    const float* __restrict__ h, const int* __restrict__ y,
    const float* __restrict__ mask, const float* __restrict__ trans,
    float* __restrict__ gold_out)
{
  const int lane = threadIdx.x & 31;
  const int b    = blockIdx.x * 8 + (threadIdx.x >> 5);
  const int*   yb = y    + (size_t)b * T_LEN;
  const float* mb = mask + (size_t)b * T_LEN;
  const float* hb = h    + (size_t)b * T_LEN * K_TAGS;

  float sc = 0.0f, lenf = 0.0f;
  for (int t = lane; t < T_LEN; t += 32) {
    const float mt = mb[t];
    const int yt = yb[t];
    const int yp = (t == 0) ? START_ID : yb[t - 1];
    sc   += (hb[t * K_TAGS + yt] + trans[yt * K_TAGS + yp]) * mt;
    lenf += mt;
  }
#pragma unroll
  for (int o = 16; o > 0; o >>= 1) {
    sc   += __shfl_down(sc,   o, 32);
    lenf += __shfl_down(lenf, o, 32);
  }
  if (lane == 0) {
    const int len  = (int)(lenf + 0.5f);   // lengths >= 1
    const int last = yb[len - 1];
    gold_out[b] = sc + trans[STOP_ID * K_TAGS + last];
  }
}

// ---------------------------------------------------------------------------
// Deterministic mean of (fwd - gold) over the batch.
// ---------------------------------------------------------------------------
__global__ __launch_bounds__(256) void crf_reduce_kernel(
    const float* __restrict__ fwd, const float* __restrict__ gold,
    float* __restrict__ out)
{
  __shared__ float sdata[256];
  float s = 0.0f;
  for (int i = threadIdx.x; i < B_BATCH; i += 256) s += fwd[i] - gold[i];
  sdata[threadIdx.x] = s;
  __syncthreads();
  for (int o = 128; o > 0; o >>= 1) {
    if (threadIdx.x < o) sdata[threadIdx.x] += sdata[threadIdx.x + o];
    __syncthreads();
  }
  if (threadIdx.x == 0) out[0] = sdata[0] / (float)B_BATCH;
}

extern "C" void kernel_launch(void* const* d_in, const int* in_sizes, int n_in,
                              void* d_out, int out_size, void* d_ws, size_t ws_size,
                              hipStream_t stream) {
  // setup_inputs() order: h (f32), y (int), mask (f32), trans (f32)
  const float* h     = (const float*)d_in[0];
  const int*   y     = (const int*)  d_in[1];
  const float* mask  = (const float*)d_in[2];
  const float* trans = (const float*)d_in[3];

  float* fwd  = (float*)d_ws;          // B floats
  float* gold = fwd + B_BATCH;         // B floats

  crf_forward_kernel<<<B_BATCH / 16, 32,  0, stream>>>(h, mask, trans, fwd);
  crf_gold_kernel   <<<B_BATCH / 8,  256, 0, stream>>>(h, y, mask, trans, gold);
  crf_reduce_kernel <<<1,            256, 0, stream>>>(fwd, gold, (float*)d_out);
}